// GNNEncoder_22780506538360
// MI455X (gfx1250) — compile-verified
//
#include <hip/hip_runtime.h>
#include <hip/hip_bf16.h>
#include <math.h>

typedef __attribute__((ext_vector_type(2))) float v2f;
typedef __attribute__((ext_vector_type(8))) float v8f;

#define THREADS 256
#define ORD_NEG_INF 0x007FFFFFu

// Monotonic order-preserving float<->uint mapping for atomicMax on floats.
__device__ __forceinline__ unsigned ford(float f) {
  unsigned u = __float_as_uint(f);
  return (u & 0x80000000u) ? ~u : (u | 0x80000000u);
}
__device__ __forceinline__ float funord(unsigned u) {
  return __uint_as_float((u & 0x80000000u) ? (u & 0x7fffffffu) : ~u);
}

// ---------------------------------------------------------------------------
// WMMA fp32 GEMM: Y[M x NT*16] = X[M x K] @ W[NT*16 x K]^T
// One wave computes a 16-row x (NT*16)-col slab with V_WMMA_F32_16X16X4_F32.
// A 16x4 layout: lanes 0-15 -> M=lane, V0=K0,V1=K1; lanes 16-31 -> K2,K3.
// C/D 16x16 layout: VGPR i -> M=i (lanes 0-15) / M=i+8 (lanes 16-31), N=lane&15.
// ---------------------------------------------------------------------------
template <int K, int NT>
__global__ void wmma_gemm_xwt(const float* __restrict__ X,
                              const float* __restrict__ W,
                              float* __restrict__ Y, int M) {
  const int NO   = NT * 16;
  const int lane = threadIdx.x & 31;
  const int wave = threadIdx.x >> 5;
  const int tile = blockIdx.x * (blockDim.x >> 5) + wave;
  if (tile * 16 >= M) return;              // wave-uniform: EXEC stays all-ones
  const int rbase = tile * 16;
  const int mrow  = rbase + (lane & 15);
  const int koff  = (lane < 16) ? 0 : 2;

  v8f acc[NT];
#pragma unroll
  for (int t = 0; t < NT; ++t) acc[t] = (v8f){0.f,0.f,0.f,0.f,0.f,0.f,0.f,0.f};

  for (int k0 = 0; k0 < K; k0 += 4) {
    const int k = k0 + koff;
    v2f a;
    a.x = (k     < K) ? X[mrow * K + k]     : 0.f;
    a.y = (k + 1 < K) ? X[mrow * K + k + 1] : 0.f;
#pragma unroll
    for (int t = 0; t < NT; ++t) {
      const int col = t * 16 + (lane & 15);
      v2f b;
      b.x = (k     < K) ? W[col * K + k]     : 0.f;
      b.y = (k + 1 < K) ? W[col * K + k + 1] : 0.f;
      acc[t] = __builtin_amdgcn_wmma_f32_16x16x4_f32(
          false, a, false, b, (short)0, acc[t], false, false);
    }
  }

  const int rowoff = (lane < 16) ? 0 : 8;
  const int colo   = lane & 15;
#pragma unroll
  for (int t = 0; t < NT; ++t) {
#pragma unroll
    for (int i = 0; i < 8; ++i) {
      Y[(rbase + rowoff + i) * NO + t * 16 + colo] = acc[t][i];
    }
  }
}

// ---------------------------------------------------------------------------
// Elementwise / scatter kernels
// ---------------------------------------------------------------------------
__global__ void fill_u32(unsigned* p, unsigned v, int n) {
  int t = blockIdx.x * blockDim.x + threadIdx.x;
  if (t < n) p[t] = v;
}

// el/er: per (node, head) dot of feat[n,h,:] with attn vectors (32 wide)
__global__ void attn_proj(const float* __restrict__ feat,
                          const float* __restrict__ al,
                          const float* __restrict__ ar,
                          float* __restrict__ el, float* __restrict__ er, int NH) {
  int t = blockIdx.x * blockDim.x + threadIdx.x;
  if (t >= NH) return;
  int n = t >> 2, h = t & 3;
  const float* f = feat + n * 128 + h * 32;
  float sl = 0.f, sr = 0.f;
#pragma unroll
  for (int i = 0; i < 32; ++i) {
    sl += f[i] * al[h * 32 + i];
    sr += f[i] * ar[h * 32 + i];
  }
  el[t] = sl; er[t] = sr;
}

__global__ void edge_score_max(const int* __restrict__ src, const int* __restrict__ dst,
                               const float* __restrict__ el, const float* __restrict__ er,
                               float* __restrict__ ebuf, unsigned* __restrict__ nmax, int EH) {
  int t = blockIdx.x * blockDim.x + threadIdx.x;
  if (t >= EH) return;
  int e = t >> 2, h = t & 3;
  float v = el[src[e] * 4 + h] + er[dst[e] * 4 + h];
  v = (v > 0.f) ? v : 0.2f * v;            // leaky_relu(., 0.2)
  ebuf[t] = v;
  atomicMax(&nmax[dst[e] * 4 + h], ford(v));
}

__global__ void edge_expsum(const int* __restrict__ dst,
                            const unsigned* __restrict__ nmax,
                            float* __restrict__ ebuf, float* __restrict__ nsum, int EH) {
  int t = blockIdx.x * blockDim.x + threadIdx.x;
  if (t >= EH) return;
  int e = t >> 2, h = t & 3;
  float m = funord(nmax[dst[e] * 4 + h]);
  float ex = expf(ebuf[t] - m);
  ebuf[t] = ex;
  atomicAdd(&nsum[dst[e] * 4 + h], ex);
}

__global__ void edge_aggregate(const int* __restrict__ src, const int* __restrict__ dst,
                               const float* __restrict__ ebuf, const float* __restrict__ nsum,
                               const float* __restrict__ feat, float* __restrict__ accum, int EH) {
  int t = blockIdx.x * blockDim.x + threadIdx.x;
  if (t >= EH) return;
  int e = t >> 2, h = t & 3;
  int s = src[e], d = dst[e];
  float a = ebuf[t] / nsum[d * 4 + h];
  const float* fs = feat + s * 128 + h * 32;
  float* ad = accum + d * 128 + h * 32;
#pragma unroll
  for (int i = 0; i < 32; ++i) atomicAdd(&ad[i], fs[i] * a);
}

// h1 = elu(rst + res1 + bias1), written in place over the accumulator
__global__ void elu_combine(float* __restrict__ accum, const float* __restrict__ res,
                            const float* __restrict__ bias, int n) {
  int t = blockIdx.x * blockDim.x + threadIdx.x;
  if (t >= n) return;
  float v = accum[t] + res[t] + bias[t & 127];
  accum[t] = (v > 0.f) ? v : expm1f(v);
}

// hnode[n,d] = mean_h( rst2[n,h,d] + h1[n,h,d] + bias2[h,d] )
__global__ void combine_mean(const float* __restrict__ accum, const float* __restrict__ h1,
                             const float* __restrict__ bias2, float* __restrict__ hnode, int ND) {
  int t = blockIdx.x * blockDim.x + threadIdx.x;
  if (t >= ND) return;
  int n = t >> 5, d = t & 31;
  float s = 0.f;
#pragma unroll
  for (int h = 0; h < 4; ++h) {
    int idx = n * 128 + h * 32 + d;
    s += accum[idx] + h1[idx] + bias2[h * 32 + d];
  }
  hnode[t] = s * 0.25f;
}

__global__ void node_weight(const float* __restrict__ hnode, const float* __restrict__ aw_w,
                            const float* __restrict__ aw_b, float* __restrict__ wbuf, int N) {
  int n = blockIdx.x * blockDim.x + threadIdx.x;
  if (n >= N) return;
  float s = aw_b[0];
#pragma unroll
  for (int i = 0; i < 32; ++i) s += hnode[n * 32 + i] * aw_w[i];
  wbuf[n] = 1.f / (1.f + expf(-s));
}

// gmat[g, 0:32] = 0 (sum half), gmat[g, 32:64] = ordered(-inf) (max half)
__global__ void graph_init(unsigned* gmat, int n) {
  int t = blockIdx.x * blockDim.x + threadIdx.x;
  if (t >= n) return;
  gmat[t] = ((t & 63) < 32) ? 0u : ORD_NEG_INF;
}

__global__ void graph_scatter(const float* __restrict__ hnode, const float* __restrict__ wbuf,
                              const int* __restrict__ gid, float* __restrict__ gmat, int ND) {
  int t = blockIdx.x * blockDim.x + threadIdx.x;
  if (t >= ND) return;
  int n = t >> 5, d = t & 31;
  int g = gid[n];
  float hv = hnode[t];
  atomicAdd(&gmat[g * 64 + d], wbuf[n] * hv);
  atomicMax((unsigned*)gmat + g * 64 + 32 + d, ford(hv));
}

// decode ordered-uint max half; never-touched (empty graph) -> 0
__global__ void graph_decode(float* gmat, int GD) {
  int t = blockIdx.x * blockDim.x + threadIdx.x;
  if (t >= GD) return;
  int g = t >> 5, d = t & 31;
  unsigned u = ((unsigned*)gmat)[g * 64 + 32 + d];
  gmat[g * 64 + 32 + d] = (u == ORD_NEG_INF) ? 0.f : funord(u);
}

// per-graph: y = yraw + out_b, then LayerNorm (no affine) over 128
__global__ void final_ln(const float* __restrict__ ybuf, const float* __restrict__ out_b,
                         float* __restrict__ out) {
  __shared__ float sm[128];
  int g = blockIdx.x, j = threadIdx.x;
  float v = ybuf[g * 128 + j] + out_b[j];
  sm[j] = v; __syncthreads();
  for (int s = 64; s > 0; s >>= 1) { if (j < s) sm[j] += sm[j + s]; __syncthreads(); }
  float mu = sm[0] * (1.f / 128.f);
  __syncthreads();
  float dv = v - mu;
  sm[j] = dv * dv; __syncthreads();
  for (int s = 64; s > 0; s >>= 1) { if (j < s) sm[j] += sm[j + s]; __syncthreads(); }
  float var = sm[0] * (1.f / 128.f);
  out[g * 128 + j] = dv * rsqrtf(var + 1e-5f);
}

// ---------------------------------------------------------------------------
extern "C" void kernel_launch(void* const* d_in, const int* in_sizes, int n_in,
                              void* d_out, int out_size, void* d_ws, size_t ws_size,
                              hipStream_t stream) {
  const float* node  = (const float*)d_in[0];
  const int*   src   = (const int*)d_in[1];
  const int*   dst   = (const int*)d_in[2];
  const int*   gid   = (const int*)d_in[3];
  const float* fc1_w = (const float*)d_in[4];
  const float* al1   = (const float*)d_in[5];
  const float* ar1   = (const float*)d_in[6];
  const float* res1w = (const float*)d_in[7];
  const float* bias1 = (const float*)d_in[8];
  const float* fc2_w = (const float*)d_in[9];
  const float* al2   = (const float*)d_in[10];
  const float* ar2   = (const float*)d_in[11];
  const float* bias2 = (const float*)d_in[12];
  const float* aw_w  = (const float*)d_in[13];
  const float* aw_b  = (const float*)d_in[14];
  const float* out_w = (const float*)d_in[15];
  const float* out_b = (const float*)d_in[16];
  (void)n_in; (void)ws_size;

  const int N = in_sizes[3];       // graph_ids has one entry per node
  const int E = in_sizes[1];
  const int G = out_size / 128;

  char* wsb = (char*)d_ws;
  size_t off = 0;
  auto alloc = [&](size_t bytes) -> char* {
    char* p = wsb + off;
    off = (off + bytes + 255) & ~(size_t)255;
    return p;
  };
  float*    bufA  = (float*)alloc((size_t)N * 128 * 4);   // feat1 / feat2
  float*    bufB  = (float*)alloc((size_t)N * 128 * 4);   // res1  / rst2 accum
  float*    bufC  = (float*)alloc((size_t)N * 128 * 4);   // rst1 accum -> h1
  float*    el    = (float*)alloc((size_t)N * 4 * 4);
  float*    er    = (float*)alloc((size_t)N * 4 * 4);
  unsigned* nmax  = (unsigned*)alloc((size_t)N * 4 * 4);
  float*    nsum  = (float*)alloc((size_t)N * 4 * 4);
  float*    ebuf  = (float*)alloc((size_t)E * 4 * 4);
  float*    hnode = (float*)alloc((size_t)N * 32 * 4);
  float*    wbuf  = (float*)alloc((size_t)N * 4);
  float*    gmat  = (float*)alloc((size_t)G * 64 * 4);
  float*    ybuf  = (float*)alloc((size_t)G * 128 * 4);

  auto grid = [](int n) { return dim3((unsigned)((n + THREADS - 1) / THREADS)); };
  const int ntiles  = (N + 15) / 16;
  const int nblocks = (ntiles + 3) / 4;                   // 4 waves / block
  const int gtiles  = (G + 15) / 16;
  const int gblocks = (gtiles + 3) / 4;

  // ---- layer 1 ----
  wmma_gemm_xwt<27, 8><<<nblocks, 128, 0, stream>>>(node, fc1_w, bufA, N);
  wmma_gemm_xwt<27, 8><<<nblocks, 128, 0, stream>>>(node, res1w, bufB, N);
  fill_u32<<<grid(N * 128), THREADS, 0, stream>>>((unsigned*)bufC, 0u, N * 128);
  fill_u32<<<grid(N * 4), THREADS, 0, stream>>>(nmax, ORD_NEG_INF, N * 4);
  fill_u32<<<grid(N * 4), THREADS, 0, stream>>>((unsigned*)nsum, 0u, N * 4);
  attn_proj<<<grid(N * 4), THREADS, 0, stream>>>(bufA, al1, ar1, el, er, N * 4);
  edge_score_max<<<grid(E * 4), THREADS, 0, stream>>>(src, dst, el, er, ebuf, nmax, E * 4);
  edge_expsum<<<grid(E * 4), THREADS, 0, stream>>>(dst, nmax, ebuf, nsum, E * 4);
  edge_aggregate<<<grid(E * 4), THREADS, 0, stream>>>(src, dst, ebuf, nsum, bufA, bufC, E * 4);
  elu_combine<<<grid(N * 128), THREADS, 0, stream>>>(bufC, bufB, bias1, N * 128);

  // ---- layer 2 ----
  wmma_gemm_xwt<128, 8><<<nblocks, 128, 0, stream>>>(bufC, fc2_w, bufA, N);
  fill_u32<<<grid(N * 128), THREADS, 0, stream>>>((unsigned*)bufB, 0u, N * 128);
  fill_u32<<<grid(N * 4), THREADS, 0, stream>>>(nmax, ORD_NEG_INF, N * 4);
  fill_u32<<<grid(N * 4), THREADS, 0, stream>>>((unsigned*)nsum, 0u, N * 4);
  attn_proj<<<grid(N * 4), THREADS, 0, stream>>>(bufA, al2, ar2, el, er, N * 4);
  edge_score_max<<<grid(E * 4), THREADS, 0, stream>>>(src, dst, el, er, ebuf, nmax, E * 4);
  edge_expsum<<<grid(E * 4), THREADS, 0, stream>>>(dst, nmax, ebuf, nsum, E * 4);
  edge_aggregate<<<grid(E * 4), THREADS, 0, stream>>>(src, dst, ebuf, nsum, bufA, bufB, E * 4);
  combine_mean<<<grid(N * 32), THREADS, 0, stream>>>(bufB, bufC, bias2, hnode, N * 32);

  // ---- readout + head ----
  node_weight<<<grid(N), THREADS, 0, stream>>>(hnode, aw_w, aw_b, wbuf, N);
  graph_init<<<grid(G * 64), THREADS, 0, stream>>>((unsigned*)gmat, G * 64);
  graph_scatter<<<grid(N * 32), THREADS, 0, stream>>>(hnode, wbuf, gid, gmat, N * 32);
  graph_decode<<<grid(G * 32), THREADS, 0, stream>>>(gmat, G * 32);
  wmma_gemm_xwt<64, 8><<<gblocks, 128, 0, stream>>>(gmat, out_w, ybuf, G);
  final_ln<<<G, 128, 0, stream>>>(ybuf, out_b, (float*)d_out);
}